// RK4Propagation_64476049047553
// MI455X (gfx1250) — compile-verified
//
#include <hip/hip_runtime.h>
#include <stdint.h>

// ---------------------------------------------------------------------------
// RK4 propagation of r' = -S^2 (mask * r), S = DAD-normalized SpMM.
// N=100000 nodes, D=64 features, E=1.6M edges, 5 RK4 steps => 40 SpMMs.
// Working set is L2-resident (x/out 25.6MB each, packed edges 25.6MB, all
// << 192MB L2): execution is L2-bandwidth/atomic bound. Data-movement plan:
//  - edges packed once into 16B records {row, col, w, pad} -> one stream
//  - per-tile staging via GLOBAL_LOAD_ASYNC_TO_LDS_B128 (ASYNCcnt path)
//  - wave32 per edge: lanes cover the 64-wide feature row as float2 loads
//  - scatter via no-return global_atomic_add_f32 (STOREcnt, L2 atomic units)
//  - global_prefetch_b8 streams the next edge tile
// ---------------------------------------------------------------------------

#define FEAT 64
#define TILE 256  // edges per block; 8 wave32s, one wave owns 32 edges

#if defined(__gfx1250__) && __has_builtin(__builtin_amdgcn_global_load_async_to_lds_b32)
#define USE_ASYNC_LDS 1
#endif

#ifdef USE_ASYNC_LDS
// AS-qualified pointer types; clang prints AS1 as __device__, AS3 as __shared__.
typedef __attribute__((address_space(1))) int gas_int;
typedef __attribute__((address_space(3))) int las_int;

#if __has_builtin(__builtin_amdgcn_global_load_async_to_lds_b128)
#define ASYNC_B128 1
// b128 form takes pointers to a 16B int vector (per clang diagnostic).
typedef int v4i_t __attribute__((vector_size(4 * sizeof(int))));
typedef __attribute__((address_space(1))) v4i_t gas_v4i;
typedef __attribute__((address_space(3))) v4i_t las_v4i;

__device__ __forceinline__ void async_copy_b128(const void* gsrc, void* ldst) {
  __builtin_amdgcn_global_load_async_to_lds_b128(
      (gas_v4i*)(uintptr_t)gsrc, (las_v4i*)(uint32_t)(uintptr_t)ldst, 0, 0);
}
#endif

__device__ __forceinline__ void async_copy_b32(const void* gsrc, void* ldst) {
  __builtin_amdgcn_global_load_async_to_lds_b32(
      (gas_int*)(uintptr_t)gsrc, (las_int*)(uint32_t)(uintptr_t)ldst, 0, 0);
}

__device__ __forceinline__ void async_wait0() {
#if __has_builtin(__builtin_amdgcn_s_wait_asynccnt)
  __builtin_amdgcn_s_wait_asynccnt(0);
#else
  asm volatile("s_wait_asynccnt 0" ::: "memory");
#endif
}
#endif  // USE_ASYNC_LDS

// -------------------- SpMM: out[rec.row] += rec.w * x[rec.col] -------------
// Block stages TILE packed 16B edge records into LDS with one async b128 per
// thread, then each wave32 walks 32 edges; lanes cover features as float2.
__global__ __launch_bounds__(256) void spmm_kernel(
    const int4* __restrict__ edges, const float* __restrict__ x,
    float* __restrict__ out, int E) {
  __shared__ int4 s_edge[TILE];  // {row, col, float_bits(w), 0}

  const int tid  = threadIdx.x;
  const int base = blockIdx.x * TILE;
  const int e    = base + tid;
  const bool valid = (e < E);

  // Stream the next edge tile toward the caches (global_prefetch_b8);
  // one prefetch per 64B cacheline (4 records).
  if (((tid & 3) == 0) && (base + TILE + tid < E)) {
    __builtin_prefetch(&edges[e + TILE], 0, 0);
  }

#if defined(USE_ASYNC_LDS) && defined(ASYNC_B128)
  if (valid) {
    async_copy_b128(edges + e, s_edge + tid);  // global_load_async_to_lds_b128
  } else {
    s_edge[tid] = make_int4(0, 0, 0, 0);       // w = 0 -> skipped below
  }
  async_wait0();                               // s_wait_asynccnt 0
  __syncthreads();
#elif defined(USE_ASYNC_LDS)
  if (valid) {
    async_copy_b32(&edges[e].x, &s_edge[tid].x);
    async_copy_b32(&edges[e].y, &s_edge[tid].y);
    async_copy_b32(&edges[e].z, &s_edge[tid].z);
    s_edge[tid].w = 0;
  } else {
    s_edge[tid] = make_int4(0, 0, 0, 0);
  }
  async_wait0();
  __syncthreads();
#else
  int4 rec = make_int4(0, 0, 0, 0);
  if (valid) {
    rec = edges[e];
    __builtin_prefetch(x + ((size_t)rec.y << 6), 0, 0);  // warm gathered row
  }
  s_edge[tid] = rec;
  __syncthreads();
#endif

  const int wave = tid >> 5;  // wave32: 8 waves per block
  const int lane = tid & 31;
  const int f    = lane << 1; // each lane owns 2 consecutive floats of 64

#pragma unroll 4
  for (int i = 0; i < 32; ++i) {
    const int4 ed = s_edge[(wave << 5) | i];   // ds_load_b128, wave-uniform
    const float wgt = __int_as_float(ed.z);
    if (wgt != 0.0f) {                         // uniform branch; skips OOB tail
      const float2 xv = *(const float2*)(x + ((size_t)ed.y << 6) + f);
      float* dst = out + ((size_t)ed.x << 6) + f;
      atomicAdd(dst,     wgt * xv.x);          // global_atomic_add_f32
      atomicAdd(dst + 1, wgt * xv.y);
    }
  }
}

// -------------------- degree accumulation (segment_sum of ones) ------------
__global__ __launch_bounds__(256) void degree_kernel(
    const int* __restrict__ row, const int* __restrict__ col,
    float* __restrict__ deg_r, float* __restrict__ deg_c, int E) {
  int e = blockIdx.x * blockDim.x + threadIdx.x;
  if (e < E) {
    atomicAdd(&deg_r[row[e]], 1.0f);
    atomicAdd(&deg_c[col[e]], 1.0f);
  }
}

// -------------------- deg -> deg^-1/2 (in place), matching reference -------
__global__ __launch_bounds__(256) void dinv_kernel(float* __restrict__ d, int n) {
  int i = blockIdx.x * blockDim.x + threadIdx.x;
  if (i < n) {
    float v = d[i];
    d[i] = (v > 0.0f) ? rsqrtf(fmaxf(v, 1.0f)) : 0.0f;
  }
}

// ---- pack edges: {row, col, w = dinv_r[row]*dinv_c[col], 0} as int4 -------
__global__ __launch_bounds__(256) void pack_kernel(
    const int* __restrict__ row, const int* __restrict__ col,
    const float* __restrict__ dinv_r, const float* __restrict__ dinv_c,
    int4* __restrict__ edges, int E) {
  int e = blockIdx.x * blockDim.x + threadIdx.x;
  if (e < E) {
    int r = row[e];
    int c = col[e];
    float wv = dinv_r[r] * dinv_c[c];
    edges[e] = make_int4(r, c, __float_as_int(wv), 0);
  }
}

// -------------------- RK4 bookkeeping (float4-vectorized) ------------------
// mask_init: xin = mask*r ; racc = r          (start of an RK4 step)
__global__ __launch_bounds__(256) void mask_init_kernel(
    float4* __restrict__ xin, float4* __restrict__ racc,
    const float4* __restrict__ r, const int* __restrict__ mask, int n4) {
  int i = blockIdx.x * blockDim.x + threadIdx.x;
  if (i < n4) {
    float4 rv = r[i];
    racc[i] = rv;
    float m = (float)mask[i >> 4];  // 16 float4 per 64-wide node row
    float4 o;
    o.x = rv.x * m; o.y = rv.y * m; o.z = rv.z * m; o.w = rv.w * m;
    xin[i] = o;
  }
}

// stage_update: racc += beta*s ; xin = mask*(r + alpha*s)
// (k_i = -s_i is folded into alpha/beta signs)
__global__ __launch_bounds__(256) void stage_update_kernel(
    float4* __restrict__ xin, float4* __restrict__ racc,
    const float4* __restrict__ r, const float4* __restrict__ s,
    const int* __restrict__ mask, float alpha, float beta, int n4) {
  int i = blockIdx.x * blockDim.x + threadIdx.x;
  if (i < n4) {
    float4 rv = r[i];
    float4 sv = s[i];
    float4 av = racc[i];
    av.x += beta * sv.x; av.y += beta * sv.y;
    av.z += beta * sv.z; av.w += beta * sv.w;
    racc[i] = av;
    float m = (float)mask[i >> 4];
    float4 o;
    o.x = m * (rv.x + alpha * sv.x);
    o.y = m * (rv.y + alpha * sv.y);
    o.z = m * (rv.z + alpha * sv.z);
    o.w = m * (rv.w + alpha * sv.w);
    xin[i] = o;
  }
}

// ---------------------------------------------------------------------------
extern "C" void kernel_launch(void* const* d_in, const int* in_sizes, int n_in,
                              void* d_out, int out_size, void* d_ws, size_t ws_size,
                              hipStream_t stream) {
  const float* r0         = (const float*)d_in[0];
  const int*   edge_index = (const int*)d_in[1];
  const int*   train_mask = (const int*)d_in[2];

  const int ND = in_sizes[0];          // N*D
  const int N  = ND / FEAT;
  const int E  = in_sizes[1] / 2;
  const int* row = edge_index;         // edge_index[0]
  const int* col = edge_index + E;     // edge_index[1]

  // --- workspace partition (256B-aligned slices), ~103MB total ------------
  float* ws  = (float*)d_ws;
  size_t off = 0;
  auto alloc = [&](size_t n) {
    float* p = ws + off;
    off += (n + 63) & ~(size_t)63;
    return p;
  };
  float* deg_r = alloc(N);
  float* deg_c = alloc(N);
  int4*  edges = (int4*)alloc((size_t)E * 4);  // packed 16B edge records
  float* xin   = alloc(ND);            // masked stage input
  float* t1    = alloc(ND);            // first-hop result
  float* sbuf  = alloc(ND);            // second-hop result s_i
  (void)ws_size; (void)n_in; (void)out_size;

  const int blocksE = (E + 255) / 256;
  const int blocksN = (N + 255) / 256;
  const int n4      = ND / 4;
  const int blocks4 = (n4 + 255) / 256;

  // --- edge weights: w = deg_r^-1/2[row] * deg_c^-1/2[col], packed --------
  (void)hipMemsetAsync(deg_r, 0, (size_t)N * sizeof(float), stream);
  (void)hipMemsetAsync(deg_c, 0, (size_t)N * sizeof(float), stream);
  degree_kernel<<<blocksE, 256, 0, stream>>>(row, col, deg_r, deg_c, E);
  dinv_kernel<<<blocksN, 256, 0, stream>>>(deg_r, N);
  dinv_kernel<<<blocksN, 256, 0, stream>>>(deg_c, N);
  pack_kernel<<<blocksE, 256, 0, stream>>>(row, col, deg_r, deg_c, edges, E);

  // --- outs[0] = r0 --------------------------------------------------------
  float* out = (float*)d_out;
  (void)hipMemcpyAsync(out, r0, (size_t)ND * sizeof(float),
                       hipMemcpyDeviceToDevice, stream);

  // --- RK4: r_{n+1} = r_n - (DT/6)(s1 + 2 s2 + 2 s3 + s4), s_i = S^2(M u_i)
  const float DT = 0.2f;
  const int   STEPS = 5;
  const float alphas[4] = { -0.5f * DT, -0.5f * DT, -DT, 0.0f };
  const float betas[4]  = { -DT / 6.0f, -DT / 3.0f, -DT / 3.0f, -DT / 6.0f };

  for (int step = 0; step < STEPS; ++step) {
    const float* rcur  = out + (size_t)step * ND;
    float*       rnext = out + (size_t)(step + 1) * ND;  // accumulator

    mask_init_kernel<<<blocks4, 256, 0, stream>>>(
        (float4*)xin, (float4*)rnext, (const float4*)rcur, train_mask, n4);

    for (int st = 0; st < 4; ++st) {
      (void)hipMemsetAsync(t1, 0, (size_t)ND * sizeof(float), stream);
      spmm_kernel<<<blocksE, 256, 0, stream>>>(edges, xin, t1, E);
      (void)hipMemsetAsync(sbuf, 0, (size_t)ND * sizeof(float), stream);
      spmm_kernel<<<blocksE, 256, 0, stream>>>(edges, t1, sbuf, E);
      stage_update_kernel<<<blocks4, 256, 0, stream>>>(
          (float4*)xin, (float4*)rnext, (const float4*)rcur,
          (const float4*)sbuf, train_mask, alphas[st], betas[st], n4);
    }
  }
}